// Conv2d_58686433132638
// MI455X (gfx1250) — compile-verified
//
#include <hip/hip_runtime.h>
#include <hip/hip_bf16.h>

typedef float v2f __attribute__((ext_vector_type(2)));
typedef float v8f __attribute__((ext_vector_type(8)));

#define H      8192   // input side
#define OH     8188   // output side (H - 5 + 1)
#define TILESX 512    // ceil(OH / 16) tiles in x
#define TY     8      // tiles per wave in y (strip height = 128 rows)

// One wave32 computes a 16-wide x 128-tall output strip as 8 vertical 16x16
// tiles, 25 x V_WMMA_F32_16X16X4_F32 per tile (banded-GEMM formulation of the
// 5x5 valid cross-correlation). B matrices / weights / column bases are
// loop-invariant; per tile all 25 A slabs are loaded into registers first so
// the full load batch is in flight before the WMMA chain consumes it.
__global__ __launch_bounds__(256) void conv5x5_wmma_f32(
    const float* __restrict__ X, const float* __restrict__ Wt,
    const float* __restrict__ Bs, float* __restrict__ Out)
{
    const int lane = threadIdx.x & 31;
    const int wib  = threadIdx.x >> 5;   // wave in block: 0..7
    const int n    = lane & 15;          // output col within tile (also A row M)
    const int kh   = lane >> 4;          // lane half selects K pair / C row half

    const int tileX = blockIdx.x * 8 + wib;   // 8 adjacent x-tiles per block
    const int x0    = tileX * 16;
    const int yBase = blockIdx.y * (16 * TY); // strip of TY tiles in y

    // ---- loop-invariant setup (once per wave) ----
    float w[25];                         // uniform -> scalar loads
#pragma unroll
    for (int t = 0; t < 25; ++t) w[t] = Wt[t];
    const float bias = Bs[0];

    // 25 banded B matrices (4x16 f32 = v2f per lane).
    // B[kg][n] = w(i, kg - n) if 0 <= kg-n < 5 else 0, kg = 4c + (p + 2*kh).
    v2f Bm[25];
#pragma unroll
    for (int i = 0; i < 5; ++i) {
#pragma unroll
        for (int c = 0; c < 5; ++c) {
            const int d0 = 4 * c + 2 * kh - n;   // p=0 tap index; p=1 is d0+1
            float b0 = 0.0f, b1 = 0.0f;
#pragma unroll
            for (int j = 0; j < 5; ++j) {        // cndmask chain, const indices
                b0 = (d0     == j) ? w[i * 5 + j] : b0;
                b1 = (d0 + 1 == j) ? w[i * 5 + j] : b1;
            }
            v2f b; b.x = b0; b.y = b1;
            Bm[i * 5 + c] = b;
        }
    }

    // Column bases per K slab (invariant over the strip); always even -> the
    // float2 A loads stay 8B aligned; clamp keeps reads in-bounds (only
    // corrupts contributions to outputs >= OH, which are store-guarded).
    unsigned colB[5];
#pragma unroll
    for (int c = 0; c < 5; ++c) {
        int cc = x0 + 4 * c + 2 * kh;
        cc = cc > (H - 2) ? (H - 2) : cc;
        colB[c] = (unsigned)cc;
    }
    const int ox = x0 + n;               // output column (fixed for the strip)

    // ---- strip loop: 8 tiles down ----
#pragma unroll 1
    for (int ty = 0; ty < TY; ++ty) {
        const int y0 = yBase + ty * 16;

        // Clamped row offsets for this tile (A row m = n reads X row y0+i+m).
        unsigned rowOff[5];
#pragma unroll
        for (int i = 0; i < 5; ++i) {
            int r = y0 + i + n;
            r = r > (H - 1) ? (H - 1) : r;
            rowOff[i] = (unsigned)r << 13;        // * 8192 elements
        }

        // Batch all 25 A-slab loads (max loads in flight, distinct dest regs).
        v2f a[25];
#pragma unroll
        for (int i = 0; i < 5; ++i) {
#pragma unroll
            for (int c = 0; c < 5; ++c) {
                a[i * 5 + c] = *(const v2f*)(X + (rowOff[i] + colB[c]));
            }
        }

        // Prefetch next tile's rows (input streams with a 16-row stride; the
        // whole input (268MB) exceeds L2 (192MB)). Both ends of the 20-col
        // span, since tile columns are only 64B aligned. Speculative-safe.
        if (ty + 1 < TY) {
#pragma unroll
            for (int i = 0; i < 5; ++i) {
                int r = y0 + 16 + i + n;
                r = r > (H - 1) ? (H - 1) : r;
                const unsigned ro = (unsigned)r << 13;
                __builtin_prefetch(X + (ro + colB[0]), 0, 1);
                __builtin_prefetch(X + (ro + colB[4]), 0, 1);
            }
        }

        // 25-deep WMMA accumulation chain.
        v8f acc = {};
#pragma unroll
        for (int i = 0; i < 5; ++i) {
#pragma unroll
            for (int c = 0; c < 5; ++c) {
                acc = __builtin_amdgcn_wmma_f32_16x16x4_f32(
                    false, a[i * 5 + c], false, Bm[i * 5 + c],
                    (short)0, acc, false, false);
            }
        }

        // C/D layout: VGPR r holds output row (r + 8*kh), column n.
#pragma unroll
        for (int r = 0; r < 8; ++r) {
            const int oy = y0 + r + 8 * kh;
            if (oy < OH && ox < OH)
                Out[(unsigned)oy * (unsigned)OH + (unsigned)ox] = acc[r] + bias;
        }
    }
}

extern "C" void kernel_launch(void* const* d_in, const int* in_sizes, int n_in,
                              void* d_out, int out_size, void* d_ws, size_t ws_size,
                              hipStream_t stream) {
    const float* X  = (const float*)d_in[0];   // 8192*8192 f32
    const float* Wt = (const float*)d_in[1];   // 5*5 f32
    const float* Bs = (const float*)d_in[2];   // 1 f32
    float* Out = (float*)d_out;                // 8188*8188 f32

    // 512 x-tiles (8 per block) x 64 y-strips of 8 tiles each.
    dim3 grid(TILESX / 8, (TILESX + TY - 1) / TY);
    dim3 block(256);
    conv5x5_wmma_f32<<<grid, block, 0, stream>>>(X, Wt, Bs, Out);
}